// HashEncoder_7954279432616
// MI455X (gfx1250) — compile-verified
//
#include <hip/hip_runtime.h>
#include <stdint.h>

#define NPTS   200000
#define NLVL   16
#define TLOG2  19
#define HMASK  ((1u << TLOG2) - 1u)
#define P1     2654435761u
#define P2     805459861u

// floor(16 * (512/16/15)^l) in fp32, l = 0..15
__constant__ float kRes[NLVL] = {
    16.f,     34.f,     72.f,     155.f,    331.f,    706.f,
    1508.f,   3217.f,   6864.f,   14643.f,  31239.f,  66644.f,
    142174.f, 303305.f, 647051.f, 1380375.f
};

__device__ __forceinline__ float lerpf(float a, float b, float w) {
    return fmaf(w, b - a, a);
}

__global__ __launch_bounds__(256) void hashenc_kernel(
    const float* __restrict__ x,        // (B,3)
    const float* __restrict__ tables,   // (L,T,2)
    const float* __restrict__ box_min,  // (3,)
    const float* __restrict__ box_max,  // (3,)
    float* __restrict__ out,            // (B, 32) -> float2 per (b,l)
    float* __restrict__ mask)           // (B,)
{
    // 16 points per block, 16 levels per point, 256 threads.
    __shared__ float s[64];   // [0..47]=x of 16 pts, [48..50]=box_min, [51..53]=box_max
    const int tid = threadIdx.x;
    const int blockPt = blockIdx.x * 16;

    // ---- CDNA5 async global->LDS staging (ASYNCcnt path) ----
    if (tid < 54) {
        const float* src;
        if (tid < 48)      src = x + (size_t)blockPt * 3 + tid;
        else if (tid < 51) src = box_min + (tid - 48);
        else               src = box_max + (tid - 51);
        uint32_t lds_addr = (uint32_t)(uintptr_t)(&s[tid]);
        asm volatile("global_load_async_to_lds_b32 %0, %1, off"
                     :: "v"(lds_addr), "v"(src)
                     : "memory");
    }
#if __has_builtin(__builtin_amdgcn_s_wait_asynccnt)
    __builtin_amdgcn_s_wait_asynccnt(0);
#else
    asm volatile("s_wait_asynccnt 0" ::: "memory");
#endif
    __syncthreads();

    const uint32_t g  = blockIdx.x * 256u + (uint32_t)tid;
    const uint32_t b  = g >> 4;          // point index
    const uint32_t l  = g & 15u;         // level index
    const int      pl = tid >> 4;        // point-in-block

    const float x0 = s[pl * 3 + 0], x1 = s[pl * 3 + 1], x2 = s[pl * 3 + 2];
    const float bn0 = s[48], bn1 = s[49], bn2 = s[50];
    const float bx0 = s[51], bx1 = s[52], bx2 = s[53];

    const float res  = kRes[l];
    const float rres = __builtin_amdgcn_rcpf(res);          // 1/res (TRANS op)

    // grid = (box_max-box_min)/res ; invgrid = res/(box_max-box_min)
    const float d0 = bx0 - bn0, d1 = bx1 - bn1, d2 = bx2 - bn2;
    const float g0 = d0 * rres, g1 = d1 * rres, g2 = d2 * rres;
    const float ig0 = res * __builtin_amdgcn_rcpf(d0);
    const float ig1 = res * __builtin_amdgcn_rcpf(d1);
    const float ig2 = res * __builtin_amdgcn_rcpf(d2);

    const float xc0 = fminf(fmaxf(x0, bn0), bx0);
    const float xc1 = fminf(fmaxf(x1, bn1), bx1);
    const float xc2 = fminf(fmaxf(x2, bn2), bx2);

    const float fb0 = floorf((xc0 - bn0) * ig0);
    const float fb1 = floorf((xc1 - bn1) * ig1);
    const float fb2 = floorf((xc2 - bn2) * ig2);

    // weights use unclamped x, min_v from clamped coords (matches reference)
    const float w0 = (x0 - fmaf(fb0, g0, bn0)) * ig0;
    const float w1 = (x1 - fmaf(fb1, g1, bn1)) * ig1;
    const float w2 = (x2 - fmaf(fb2, g2, bn2)) * ig2;

    const int bl0 = (int)fb0, bl1 = (int)fb1, bl2 = (int)fb2;

    const uint32_t hx0 = (uint32_t)bl0;            // * prime 1
    const uint32_t hx1 = (uint32_t)(bl0 + 1);
    const uint32_t hy0 = (uint32_t)bl1 * P1;
    const uint32_t hy1 = (uint32_t)(bl1 + 1) * P1;
    const uint32_t hz0 = (uint32_t)bl2 * P2;
    const uint32_t hz1 = (uint32_t)(bl2 + 1) * P2;

    // Fold level into one 32-bit element offset so the compiler can keep a
    // single scalar base (SADDR) + 32-bit voffset per gather. Max byte offset
    // = 2^23 * 8 = 64MB, fits easily. Level bits don't overlap hash bits.
    const uint32_t lvlbase = l << TLOG2;
    const float2* __restrict__ tabs = (const float2*)tables;

    const float2 e000 = tabs[lvlbase | ((hx0 ^ hy0 ^ hz0) & HMASK)];
    const float2 e001 = tabs[lvlbase | ((hx0 ^ hy0 ^ hz1) & HMASK)];
    const float2 e010 = tabs[lvlbase | ((hx0 ^ hy1 ^ hz0) & HMASK)];
    const float2 e011 = tabs[lvlbase | ((hx0 ^ hy1 ^ hz1) & HMASK)];
    const float2 e100 = tabs[lvlbase | ((hx1 ^ hy0 ^ hz0) & HMASK)];
    const float2 e101 = tabs[lvlbase | ((hx1 ^ hy0 ^ hz1) & HMASK)];
    const float2 e110 = tabs[lvlbase | ((hx1 ^ hy1 ^ hz0) & HMASK)];
    const float2 e111 = tabs[lvlbase | ((hx1 ^ hy1 ^ hz1) & HMASK)];

    // trilinear: x-dim pairs (0,4)(1,5)(2,6)(3,7), then y, then z
    float c00x = lerpf(e000.x, e100.x, w0), c00y = lerpf(e000.y, e100.y, w0);
    float c01x = lerpf(e001.x, e101.x, w0), c01y = lerpf(e001.y, e101.y, w0);
    float c10x = lerpf(e010.x, e110.x, w0), c10y = lerpf(e010.y, e110.y, w0);
    float c11x = lerpf(e011.x, e111.x, w0), c11y = lerpf(e011.y, e111.y, w0);

    float c0x = lerpf(c00x, c10x, w1), c0y = lerpf(c00y, c10y, w1);
    float c1x = lerpf(c01x, c11x, w1), c1y = lerpf(c01y, c11y, w1);

    float2 f;
    f.x = lerpf(c0x, c1x, w2);
    f.y = lerpf(c0y, c1y, w2);

    // out offset b*32 + l*2 == 2*g floats == g-th float2. Wave writes 256
    // contiguous bytes, fully coalesced. Output is never re-read on device:
    // non-temporal store keeps the table working set resident in L2.
    union { float2 v; double d; } pk;
    pk.v = f;
    __builtin_nontemporal_store(pk.d, (double*)out + g);

    if (l == 0) {
        bool keep = (x0 >= bn0) & (x0 <= bx0) &
                    (x1 >= bn1) & (x1 <= bx1) &
                    (x2 >= bn2) & (x2 <= bx2);
        __builtin_nontemporal_store(keep ? 1.0f : 0.0f, mask + b);
    }
}

extern "C" void kernel_launch(void* const* d_in, const int* in_sizes, int n_in,
                              void* d_out, int out_size, void* d_ws, size_t ws_size,
                              hipStream_t stream) {
    const float* x       = (const float*)d_in[0];
    const float* tables  = (const float*)d_in[1];
    const float* box_min = (const float*)d_in[2];
    const float* box_max = (const float*)d_in[3];

    float* out  = (float*)d_out;
    float* mask = out + (size_t)NPTS * (NLVL * 2);

    // B*L = 3,200,000 threads; 256/block -> 12500 blocks exactly
    const int nthreads = NPTS * NLVL;
    dim3 grid((nthreads + 255) / 256);
    hashenc_kernel<<<grid, 256, 0, stream>>>(x, tables, box_min, box_max, out, mask);
}